// RoIHeads_22213570855103
// MI455X (gfx1250) — compile-verified
//
#include <hip/hip_runtime.h>
#include <math.h>

// ---------- problem constants ----------
#define N_ROWS 4096
#define N_CLS  91
#define R_KP   1024
#define K_KP   17
#define HM     56
#define RES    64
#define NMAPS  (R_KP * K_KP)          // 17408
#define SL1_BETA (1.0f / 9.0f)

// d_out layout (float32), reference tuple flattened in return order
#define OUT_CLS  0
#define OUT_BOX  1
#define OUT_XY   2
#define OUT_ES   (OUT_XY + NMAPS * 3)         // 52226
#define OUT_MASK (OUT_ES + NMAPS)             // 69634

// d_ws layout (floats)
#define WS_PART  0            // 64 floats: 2 partial sums x 32 blocks
#define WS_W     64           // 64x56 resize weight matrix

typedef float v2f __attribute__((ext_vector_type(2)));
typedef float v8f __attribute__((ext_vector_type(8)));

// Keys cubic, a = -0.5 (jax bicubic kernel)
__device__ __forceinline__ float keys_cubic(float t) {
    t = fabsf(t);
    if (t < 1.0f) return ((1.5f * t - 2.5f) * t) * t + 1.0f;
    if (t < 2.0f) return ((-0.5f * t + 2.5f) * t - 4.0f) * t + 2.0f;
    return 0.0f;
}

// W[i][j], i in [0,64), j in [0,56): dense normalized bicubic weight row,
// matching jax.image.resize's weight-matrix construction (out-of-range taps
// dropped, row renormalized).
__global__ void init_weights(float* __restrict__ W) {
    int i = threadIdx.x;
    if (i >= RES) return;
    float x = (i + 0.5f) * ((float)HM / (float)RES) - 0.5f;
    float s = 0.0f;
    for (int j = 0; j < HM; ++j) s += keys_cubic(x - (float)j);
    float inv = 1.0f / s;
    for (int j = 0; j < HM; ++j) W[i * HM + j] = keys_cubic(x - (float)j) * inv;
}

// ---------- fastrcnn loss: one wave32 per row ----------
__global__ void __launch_bounds__(256)
loss_kernel(const float* __restrict__ logits, const float* __restrict__ boxreg,
            const int* __restrict__ labels, const float* __restrict__ tgt,
            float* __restrict__ part) {
    __shared__ float wcls[8];
    __shared__ float wbox[8];
    const int tid = threadIdx.x, wave = tid >> 5, lane = tid & 31;
    float cls_acc = 0.0f, box_acc = 0.0f;
    const int base_row = blockIdx.x * 128 + wave * 16;

    for (int it = 0; it < 16; ++it) {
        const int n = base_row + it;
        const float* row = logits + (size_t)n * N_CLS;
        // 91 cols over 32 lanes: lane, lane+32, lane+64(partial)
        float v0 = row[lane];
        float v1 = row[lane + 32];
        bool has2 = (lane + 64) < N_CLS;
        float v2 = has2 ? row[lane + 64] : -3.402823466e38f;
        float m = fmaxf(fmaxf(v0, v1), v2);
        for (int off = 16; off; off >>= 1) m = fmaxf(m, __shfl_xor(m, off, 32));
        float s = expf(v0 - m) + expf(v1 - m) + (has2 ? expf(v2 - m) : 0.0f);
        for (int off = 16; off; off >>= 1) s += __shfl_xor(s, off, 32);

        if (lane == 0) {
            int l = labels[n];
            float ll = row[l];
            cls_acc += -(ll - m - logf(s));
            float sl = 0.0f;
            for (int j = 0; j < 4; ++j) {
                float d = boxreg[(size_t)n * (N_CLS * 4) + l * 4 + j] - tgt[n * 4 + j];
                float ad = fabsf(d);
                sl += (ad < SL1_BETA) ? (0.5f * d * d / SL1_BETA) : (ad - 0.5f * SL1_BETA);
            }
            if (l > 0) box_acc += sl;
        }
    }
    if (lane == 0) { wcls[wave] = cls_acc; wbox[wave] = box_acc; }
    __syncthreads();
    if (tid == 0) {
        float c = 0.0f, b = 0.0f;
        for (int w = 0; w < 8; ++w) { c += wcls[w]; b += wbox[w]; }
        part[blockIdx.x * 2 + 0] = c;
        part[blockIdx.x * 2 + 1] = b;
    }
}

__global__ void finalize_loss(const float* __restrict__ part, float* __restrict__ out) {
    if (blockIdx.x == 0 && threadIdx.x == 0) {
        float c = 0.0f, b = 0.0f;
        for (int i = 0; i < 32; ++i) { c += part[2 * i]; b += part[2 * i + 1]; }
        out[OUT_CLS] = c / (float)N_ROWS;
        out[OUT_BOX] = b / (float)N_ROWS;
    }
}

// ---------- keypoint: bicubic resize via fp32 WMMA + argmax ----------
// One block (256 thr = 8 wave32) per (r,k) map.
//   stage1: T(64x64)  = in_pad(64x64) * W^T      (T[m][n] = sum_k in[m][k]*W[n][k])
//   stage2: O(64x64)  = W * T                    (O[m][n] = sum_k W[m][k]*T[k][n])
// 16 tiles of 16x16 per stage, 2 tiles per wave, K-loop 14 x (K=4).
// f32 16x16x4 lane layout: lane = {h=lane>>4, r=lane&15};
//   A vgpr v -> A[r][K=2h+v]; B vgpr v -> B[K=2h+v][r]; C vgpr v -> C[M=v+8h][r].
__global__ void __launch_bounds__(256)
kp_kernel(const float* __restrict__ maps, const float* __restrict__ rois,
          const float* __restrict__ Wg, float* __restrict__ out) {
    __shared__ float in_pad[64 * 64];
    __shared__ float Tld[64 * 64];
    __shared__ float Wld[RES * HM];
    __shared__ float smax[256];
    __shared__ int   sidx[256];

    const int tid = threadIdx.x;
    const int map = blockIdx.x;

    for (int i = tid; i < 64 * 64; i += 256) in_pad[i] = 0.0f;
    for (int i = tid; i < RES * HM; i += 256) Wld[i] = Wg[i];
    __syncthreads();

    // 56x56 map -> padded LDS (row stride 64). 56 = 14 float4 per row.
    const float4* src = (const float4*)(maps + (size_t)map * (HM * HM));
    for (int f = tid; f < (HM * HM) / 4; f += 256) {
        int row = f / 14, c4 = f % 14;
        *(float4*)&in_pad[row * 64 + c4 * 4] = src[f];
    }
    __syncthreads();

    const int wave = tid >> 5, lane = tid & 31;
    const int h = lane >> 4, r = lane & 15;

    // ---- stage 1 ----
    for (int j = 0; j < 2; ++j) {
        const int tile = wave * 2 + j, mt = tile >> 2, nt = tile & 3;
        v8f acc = {0.f, 0.f, 0.f, 0.f, 0.f, 0.f, 0.f, 0.f};
        const float* arow = &in_pad[(mt * 16 + r) * 64];
        const float* brow = &Wld[(nt * 16 + r) * HM];   // B[k][n] = W[n][k]
        for (int kk = 0; kk < HM; kk += 4) {
            v2f a, b;
            a[0] = arow[kk + 2 * h];  a[1] = arow[kk + 2 * h + 1];
            b[0] = brow[kk + 2 * h];  b[1] = brow[kk + 2 * h + 1];
            acc = __builtin_amdgcn_wmma_f32_16x16x4_f32(
                false, a, false, b, (short)0, acc, false, false);
        }
        for (int vv = 0; vv < 8; ++vv)
            Tld[(mt * 16 + vv + 8 * h) * 64 + nt * 16 + r] = acc[vv];
    }
    __syncthreads();

    // ---- stage 2 + register-resident argmax ----
    float bestV = -3.402823466e38f;
    int   bestI = 0x7fffffff;
    for (int j = 0; j < 2; ++j) {
        const int tile = wave * 2 + j, mt = tile >> 2, nt = tile & 3;
        v8f acc = {0.f, 0.f, 0.f, 0.f, 0.f, 0.f, 0.f, 0.f};
        const float* arow = &Wld[(mt * 16 + r) * HM];
        for (int kk = 0; kk < HM; kk += 4) {
            v2f a, b;
            a[0] = arow[kk + 2 * h];  a[1] = arow[kk + 2 * h + 1];
            b[0] = Tld[(kk + 2 * h) * 64 + nt * 16 + r];
            b[1] = Tld[(kk + 2 * h + 1) * 64 + nt * 16 + r];
            acc = __builtin_amdgcn_wmma_f32_16x16x4_f32(
                false, a, false, b, (short)0, acc, false, false);
        }
        for (int vv = 0; vv < 8; ++vv) {
            const int y = mt * 16 + vv + 8 * h, x = nt * 16 + r;
            const float v = acc[vv];
            const int idx = y * RES + x;
            if (v > bestV || (v == bestV && idx < bestI)) { bestV = v; bestI = idx; }
        }
    }
    smax[tid] = bestV;
    sidx[tid] = bestI;
    __syncthreads();

    if (tid == 0) {
        float bv = smax[0]; int bi = sidx[0];
        for (int i = 1; i < 256; ++i)
            if (smax[i] > bv || (smax[i] == bv && sidx[i] < bi)) { bv = smax[i]; bi = sidx[i]; }
        const int rr = map / K_KP;
        const float x0 = rois[rr * 4 + 0], y0 = rois[rr * 4 + 1];
        const float w  = fmaxf(rois[rr * 4 + 2] - x0, 1.0f);
        const float ht = fmaxf(rois[rr * 4 + 3] - y0, 1.0f);
        const int xi = bi % RES, yi = bi / RES;
        float* xy = out + OUT_XY + (size_t)map * 3;
        xy[0] = ((float)xi + 0.5f) * (w  / (float)RES) + x0;
        xy[1] = ((float)yi + 0.5f) * (ht / (float)RES) + y0;
        xy[2] = 1.0f;
        out[OUT_ES + map] = bv;
    }
}

// ---------- mask: gather labeled class slice, then sigmoid ----------
__global__ void __launch_bounds__(256)
mask_kernel(const float* __restrict__ logits, const int* __restrict__ labels,
            float* __restrict__ out) {
    const int rr = blockIdx.x;
    const int c = labels[rr];
    const float* src = logits + ((size_t)rr * N_CLS + c) * 784;
    float* dst = out + OUT_MASK + (size_t)rr * 784;
    for (int i = threadIdx.x; i < 784; i += 256) {
        float v = src[i];
        dst[i] = 1.0f / (1.0f + expf(-v));
    }
}

extern "C" void kernel_launch(void* const* d_in, const int* in_sizes, int n_in,
                              void* d_out, int out_size, void* d_ws, size_t ws_size,
                              hipStream_t stream) {
    (void)in_sizes; (void)n_in; (void)out_size; (void)ws_size;
    const float* class_logits   = (const float*)d_in[0];
    const float* box_regression = (const float*)d_in[1];
    const int*   labels         = (const int*)d_in[2];
    const float* reg_targets    = (const float*)d_in[3];
    const float* kp_maps        = (const float*)d_in[4];
    const float* kp_rois        = (const float*)d_in[5];
    const float* mask_logits    = (const float*)d_in[6];
    const int*   mask_labels    = (const int*)d_in[7];
    float* out = (float*)d_out;
    float* ws  = (float*)d_ws;

    hipLaunchKernelGGL(init_weights, dim3(1), dim3(64), 0, stream, ws + WS_W);
    hipLaunchKernelGGL(loss_kernel, dim3(32), dim3(256), 0, stream,
                       class_logits, box_regression, labels, reg_targets, ws + WS_PART);
    hipLaunchKernelGGL(finalize_loss, dim3(1), dim3(32), 0, stream, ws + WS_PART, out);
    hipLaunchKernelGGL(kp_kernel, dim3(NMAPS), dim3(256), 0, stream,
                       kp_maps, kp_rois, ws + WS_W, out);
    hipLaunchKernelGGL(mask_kernel, dim3(R_KP), dim3(256), 0, stream,
                       mask_logits, mask_labels, out);
}